// _TriDirMambaOrFallback_83854941487464
// MI455X (gfx1250) — compile-verified
//
#include <hip/hip_runtime.h>
#include <hip/hip_bf16.h>

typedef float v2f __attribute__((ext_vector_type(2)));
typedef float v8f __attribute__((ext_vector_type(8)));

#define MVOX 262144   // D*H*W = 64^3
#define CCH  48
#define NST  16
#define LOG2E 1.4426950408889634f
#define LN2   0.6931471805599453f

__device__ __forceinline__ float fast_exp2(float x) { return __builtin_amdgcn_exp2f(x); }
__device__ __forceinline__ float fast_log2(float x) { return __builtin_amdgcn_logf(x); }

// softplus(z) = max(z,0) + ln(1 + e^{-|z|})
__device__ __forceinline__ float softplus_f(float z) {
    float e = fast_exp2(-fabsf(z) * LOG2E);
    return fmaxf(z, 0.f) + fast_log2(1.f + e) * LN2;
}

// ---------------------------------------------------------------------------
// Projection kernel: per-direction GEMMs via V_WMMA_F32_16X16X4_F32.
//   x   : [C=48, M] channel-major (k-columns contiguous in m -> perfect A frags)
//   dtb : [M, 48]  = softplus(x^T @ Wdt + bdt)
//   Bcb : [M, 16]  = x^T @ WB
//   Ccb : [M, 16]  = x^T @ WC
//   xT  : [M, 48]  transposed copy of x (written when write_xT != 0)
// One wave = one 16-row tile; block = 8 waves = 128 rows.
// Weights are staged fragment-ordered in LDS: each k-step B-fragment is one
// ds_load_b64 per lane instead of two global_load_b32.
// ---------------------------------------------------------------------------
__global__ __launch_bounds__(256) void tri_mamba_proj_wmma(
    const float* __restrict__ x,
    const float* __restrict__ Wdt,   // [48,48] row-major (k, n)
    const float* __restrict__ bdt,   // [48]
    const float* __restrict__ WB,    // [48,16]
    const float* __restrict__ WC,    // [48,16]
    float* __restrict__ dtb,
    float* __restrict__ Bcb,
    float* __restrict__ Ccb,
    float* __restrict__ xT,
    int write_xT)
{
    // Fragment-ordered weight staging:
    //   sWdt[((t*12 + kk)*32 + half*16 + nl)*2 + j] = Wdt[(kk*4 + 2*half + j)*48 + t*16 + nl]
    __shared__ float sWdt[3 * 12 * 64];   // 2304 floats
    __shared__ float sWB [12 * 64];       // 768 floats
    __shared__ float sWC [12 * 64];       // 768 floats

    for (int i = threadIdx.x; i < 2304; i += 256) {
        const int j  = i & 1;
        const int n_ = (i >> 1) & 15;
        const int hf = (i >> 5) & 1;
        const int r  = i >> 6;            // r = t*12 + kk
        const int kk_ = r % 12;
        const int t   = r / 12;
        sWdt[i] = Wdt[(kk_ * 4 + 2 * hf + j) * CCH + t * 16 + n_];
    }
    for (int i = threadIdx.x; i < 768; i += 256) {
        const int j  = i & 1;
        const int n_ = (i >> 1) & 15;
        const int hf = (i >> 5) & 1;
        const int kk_ = i >> 6;
        sWB[i] = WB[(kk_ * 4 + 2 * hf + j) * NST + n_];
        sWC[i] = WC[(kk_ * 4 + 2 * hf + j) * NST + n_];
    }
    __syncthreads();

    const int wave = threadIdx.x >> 5;
    const int lane = threadIdx.x & 31;
    const int half = lane >> 4;      // A: which K pair; C/D: which M half
    const int nl   = lane & 15;      // A: M row; B,C/D: N column
    const size_t m0 = (size_t)blockIdx.x * 128 + (size_t)wave * 16;
    const int fo   = (half * 16 + nl) * 2;  // lane offset inside a fragment slab

    v8f acc0 = {}, acc1 = {}, acc2 = {}, accB = {}, accC = {};

    #pragma unroll 4
    for (int kk = 0; kk < 12; ++kk) {
        const int ka = kk * 4 + 2 * half;   // this lane-half's K base
        v2f a;
        a.x = x[(size_t)ka       * MVOX + m0 + nl];
        a.y = x[(size_t)(ka + 1) * MVOX + m0 + nl];

        const v2f b0 = *(const v2f*)&sWdt[(0 * 12 + kk) * 64 + fo];
        const v2f b1 = *(const v2f*)&sWdt[(1 * 12 + kk) * 64 + fo];
        const v2f b2 = *(const v2f*)&sWdt[(2 * 12 + kk) * 64 + fo];
        const v2f bB = *(const v2f*)&sWB [kk * 64 + fo];
        const v2f bC = *(const v2f*)&sWC [kk * 64 + fo];

        acc0 = __builtin_amdgcn_wmma_f32_16x16x4_f32(false, a, false, b0, (short)0, acc0, false, false);
        acc1 = __builtin_amdgcn_wmma_f32_16x16x4_f32(false, a, false, b1, (short)0, acc1, false, false);
        acc2 = __builtin_amdgcn_wmma_f32_16x16x4_f32(false, a, false, b2, (short)0, acc2, false, false);
        accB = __builtin_amdgcn_wmma_f32_16x16x4_f32(false, a, false, bB, (short)0, accB, false, false);
        accC = __builtin_amdgcn_wmma_f32_16x16x4_f32(false, a, false, bC, (short)0, accC, false, false);
    }

    // Epilogue: bias + softplus for dt, direct store for B/C projections.
    const float bb0 = bdt[nl], bb1 = bdt[16 + nl], bb2 = bdt[32 + nl];
    #pragma unroll
    for (int v = 0; v < 8; ++v) {
        const size_t mr = m0 + (size_t)v + (size_t)(8 * half);
        dtb[mr * CCH + nl]      = softplus_f(acc0[v] + bb0);
        dtb[mr * CCH + 16 + nl] = softplus_f(acc1[v] + bb1);
        dtb[mr * CCH + 32 + nl] = softplus_f(acc2[v] + bb2);
        Bcb[mr * NST + nl] = accB[v];
        Ccb[mr * NST + nl] = accC[v];
    }

    // One-time transposed copy of x for the scan phase (coalesced reads).
    if (write_xT) {
        const size_t mb = (size_t)blockIdx.x * 128;
        for (int i = threadIdx.x; i < 128 * CCH; i += 256) {
            const int r = i & 127;
            const int c = i >> 7;
            xT[(mb + (size_t)r) * CCH + c] = x[(size_t)c * MVOX + mb + (size_t)r];
        }
    }
}

// ---------------------------------------------------------------------------
// Selective scan along one spatial axis. One thread owns one sequence and
// THREE contiguous channels (3x less redundant Bc/Cc traffic), carrying
// h[3][N=16] in registers across L=64 steps.
//   dir 0: scan D, parallel (h,w):   vox = l*4096 + seq
//   dir 1: scan H, parallel (d,w):   vox = (seq/64)*4096 + l*64 + (seq%64)
//   dir 2: scan W, parallel (d,h):   vox = seq*64 + l
// Block = 256 threads = 16 sequences x 16 channel-groups.
// ---------------------------------------------------------------------------
__global__ __launch_bounds__(256) void tri_mamba_scan(
    const float* __restrict__ xT,    // [M,48]
    const float* __restrict__ dtb,   // [M,48] (softplus applied)
    const float* __restrict__ Bcb,   // [M,16]
    const float* __restrict__ Ccb,   // [M,16]
    const float* __restrict__ A_log, // [3,48,16]
    const float* __restrict__ D_skip,// [3,48]
    float* __restrict__ yws,         // [M,48] accumulator
    int dir)
{
    const int tid = blockIdx.x * 256 + threadIdx.x;
    const int seq = tid >> 4;        // 0..4095
    const int cg  = tid & 15;        // channel group -> channels 3*cg .. 3*cg+2
    const int c0  = cg * 3;

    size_t base; size_t strideL;
    if      (dir == 0) { base = (size_t)seq;                                    strideL = 4096; }
    else if (dir == 1) { base = (size_t)(seq >> 6) * 4096 + (size_t)(seq & 63); strideL = 64; }
    else               { base = (size_t)seq * 64;                               strideL = 1; }

    // A2[j][n] = -exp(A_log[c0+j][n]) * log2(e):  decay = exp2(dt * A2)  (one v_exp_f32)
    float A2[3][NST];
    float dskip[3];
    #pragma unroll
    for (int j = 0; j < 3; ++j) {
        dskip[j] = D_skip[dir * CCH + c0 + j];
        #pragma unroll
        for (int n = 0; n < NST; ++n)
            A2[j][n] = -fast_exp2(A_log[((size_t)dir * CCH + c0 + j) * NST + n] * LOG2E) * LOG2E;
    }

    float h[3][NST];
    #pragma unroll
    for (int j = 0; j < 3; ++j)
        #pragma unroll
        for (int n = 0; n < NST; ++n) h[j][n] = 0.f;

    for (int l = 0; l < 64; ++l) {
        const size_t vox = base + (size_t)l * strideL;

        const float* dp = dtb + vox * CCH + c0;
        const float* xp = xT  + vox * CCH + c0;
        float dtv[3], xv[3];
        #pragma unroll
        for (int j = 0; j < 3; ++j) { dtv[j] = dp[j]; xv[j] = xp[j]; }

        const float4* bp4 = (const float4*)(Bcb + vox * NST);
        const float4* cp4 = (const float4*)(Ccb + vox * NST);
        float4 Bq[4], Cq[4];
        #pragma unroll
        for (int q = 0; q < 4; ++q) { Bq[q] = bp4[q]; Cq[q] = cp4[q]; }
        const float* Bv = (const float*)Bq;
        const float* Cv = (const float*)Cq;

        float y[3];
        #pragma unroll
        for (int j = 0; j < 3; ++j) {
            const float k = dtv[j] * xv[j];
            float acc = 0.f;
            #pragma unroll
            for (int n = 0; n < NST; ++n) {
                const float decay = fast_exp2(dtv[j] * A2[j][n]);
                h[j][n] = fmaf(decay, h[j][n], k * Bv[n]);
                acc     = fmaf(h[j][n], Cv[n], acc);
            }
            y[j] = fmaf(dskip[j], xv[j], acc);
        }

        float* yp = yws + vox * CCH + c0;
        if (dir == 0) {
            #pragma unroll
            for (int j = 0; j < 3; ++j) yp[j] = y[j];       // first direction initializes
        } else {
            #pragma unroll
            for (int j = 0; j < 3; ++j) yp[j] += y[j];      // later directions accumulate
        }
    }
}

// ---------------------------------------------------------------------------
// Final transpose: yws [M,48] voxel-major -> out [48,M] channel-major.
// ---------------------------------------------------------------------------
__global__ __launch_bounds__(256) void tri_mamba_writeout(
    const float* __restrict__ yws, float* __restrict__ out)
{
    const size_t m = (size_t)blockIdx.x * 256 + threadIdx.x;
    #pragma unroll 4
    for (int c = 0; c < CCH; ++c)
        out[(size_t)c * MVOX + m] = yws[m * CCH + c];
}

extern "C" void kernel_launch(void* const* d_in, const int* in_sizes, int n_in,
                              void* d_out, int out_size, void* d_ws, size_t ws_size,
                              hipStream_t stream) {
    const float* x      = (const float*)d_in[0];  // [1,48,64,64,64]
    const float* A_log  = (const float*)d_in[1];  // [3,48,16]
    const float* W_dt   = (const float*)d_in[2];  // [3,48,48]
    const float* b_dt   = (const float*)d_in[3];  // [3,48]
    const float* W_B    = (const float*)d_in[4];  // [3,48,16]
    const float* W_C    = (const float*)d_in[5];  // [3,48,16]
    const float* D_skip = (const float*)d_in[6];  // [3,48]
    float* out = (float*)d_out;

    const size_t M = MVOX;
    float* ws  = (float*)d_ws;
    float* xT  = ws;                // M*48
    float* dtb = xT  + M * CCH;     // M*48
    float* Bcb = dtb + M * CCH;     // M*16
    float* Ccb = Bcb + M * NST;     // M*16
    float* yws = Ccb + M * NST;     // M*48

    for (int d = 0; d < 3; ++d) {
        tri_mamba_proj_wmma<<<MVOX / 128, 256, 0, stream>>>(
            x,
            W_dt + (size_t)d * CCH * CCH,
            b_dt + (size_t)d * CCH,
            W_B  + (size_t)d * CCH * NST,
            W_C  + (size_t)d * CCH * NST,
            dtb, Bcb, Ccb, xT, d == 0 ? 1 : 0);

        tri_mamba_scan<<<(4096 * 16) / 256, 256, 0, stream>>>(
            xT, dtb, Bcb, Ccb, A_log, D_skip, yws, d);
    }

    tri_mamba_writeout<<<MVOX / 256, 256, 0, stream>>>(yws, out);
}